// MS_19241453486857
// MI455X (gfx1250) — compile-verified
//
#include <hip/hip_runtime.h>

typedef float v2f __attribute__((ext_vector_type(2)));
typedef float v8f __attribute__((ext_vector_type(8)));

#define EMBED_DIM 32
#define RES 96
#define NPIX (RES * RES)          // 9216
#define BW_F 2.0f
#define ETA_F 0.5f
#define N_ITERS 3
#define PANEL 128                 // i-columns staged per LDS panel
#define NPANELS (NPIX / PANEL)    // 72
#define LDS_PITCH 132             // padded row pitch (floats): row stride = 4 banks -> conflict-free column reads
#define PANEL_F (EMBED_DIM * LDS_PITCH)   // floats per buffer
#define WAVES 4
#define BLOCK (WAVES * 32)
#define JCOLS (WAVES * 16)        // 64 j-columns per block
#define LOG2E_F 1.44269504088896340736f

// lanes<16 read src for row RL from p/q; lanes>=16 read row RH.
// p[r]: lanes0-15 = row r, lanes16-31 = row r+8 (col = lane&15)
// q[r] = SWAPX16(p[r]): lanes0-15 = row r+8, lanes16-31 = row r
#define SEL_B(RL, RH) (lo ? (((RL) < 8) ? p[(RL)] : q[(RL) - 8]) \
                          : (((RH) < 8) ? q[(RH)] : p[(RH) - 8]))

__device__ __forceinline__ float swap16(float x) {
    // ds_swizzle group-of-32: xor=0x10, or=0, and=0x1f  -> swap lane halves
    return __builtin_bit_cast(float,
        __builtin_amdgcn_ds_swizzle(__builtin_bit_cast(int, x), 0x401F));
}

__global__ __launch_bounds__(BLOCK)
void ms_iter_kernel(const float* __restrict__ Xin, float* __restrict__ Xout) {
    __shared__ float lds[2 * PANEL_F];   // double-buffered panel

    const int tid   = threadIdx.x;
    const int lane  = tid & 31;
    const int wave  = tid >> 5;
    const bool lo   = lane < 16;
    const int c     = lane & 15;                    // fragment column
    const int jbase = blockIdx.x * JCOLS + wave * 16;

    // LDS byte offset of the buffers. Deriving this from the array pointer
    // (ptrtoint) makes the shared array escape into the asm, so the compiler
    // must honor the async writes and keep our ds_loads.
    const unsigned lbase = (unsigned)(size_t)(void*)lds;

    // issue one panel (32 x 128 f32 = 16 KB) as 8 async b128 loads per thread group
    auto issue_panel = [&](int ibase, int buf) {
        const float* src = Xin + ibase;
#pragma unroll
        for (int ch = 0; ch < (EMBED_DIM * PANEL * 4) / (16 * BLOCK); ++ch) {
            const unsigned f      = (unsigned)(ch * BLOCK + tid) * 16u;  // byte offset in panel
            const unsigned row    = f >> 9;                              // 512 B per panel row
            const unsigned ldsoff = lbase + (unsigned)buf * (PANEL_F * 4u)
                                  + row * (LDS_PITCH * 4u) + (f & 511u);
            const unsigned voff   = row * (NPIX * 4u) + (f & 511u);
            asm volatile("global_load_async_to_lds_b128 %0, %1, %2"
                         :: "v"(ldsoff), "v"(voff), "s"(src) : "memory");
        }
    };

    // ---- B fragment for gemm1 (resident): B1[k][n] = X[d=k][jbase+n], 8 K-chunks of 4 over D=32
    // layout: v.x = (k=4kk+0 lanes0-15 | k=4kk+2 lanes16-31), v.y = (+1 | +3)
    v2f bj[8];
#pragma unroll
    for (int kk = 0; kk < 8; ++kk) {
        const int r0 = 4 * kk + (lo ? 0 : 2);
        bj[kk].x = Xin[(r0 + 0) * NPIX + jbase + c];
        bj[kk].y = Xin[(r0 + 1) * NPIX + jbase + c];
    }

    v8f acc0 = {0.f, 0.f, 0.f, 0.f, 0.f, 0.f, 0.f, 0.f};  // Y rows d=0..15
    v8f acc1 = {0.f, 0.f, 0.f, 0.f, 0.f, 0.f, 0.f, 0.f};  // Y rows d=16..31
    float deg = 0.f;

    // ---- software pipeline: keep two async groups (8 loads each) in flight
    issue_panel(0, 0);
    issue_panel(PANEL, 1);

    for (int t = 0; t < NPANELS; ++t) {
        // retire the oldest group: async loads complete in order, so
        // asynccnt<=8 means panel t's 8 loads are done (panel t+1 may still fly)
        if (t + 1 < NPANELS) asm volatile("s_wait_asynccnt 0x8" ::: "memory");
        else                 asm volatile("s_wait_asynccnt 0x0" ::: "memory");
        __syncthreads();

        const float* L = lds + (t & 1) * PANEL_F;

#pragma unroll 1
        for (int it = 0; it < PANEL / 16; ++it) {
            const int ic = it * 16;

            // ---- gemm1: S(16i x 16j) = Xi^T * Xj, 8 WMMA f32 16x16x4 over D=32
            v8f S = {0.f, 0.f, 0.f, 0.f, 0.f, 0.f, 0.f, 0.f};
#pragma unroll
            for (int kk = 0; kk < 8; ++kk) {
                const int r0 = 4 * kk + (lo ? 0 : 2);
                v2f a;
                a.x = L[(r0 + 0) * LDS_PITCH + ic + c];
                a.y = L[(r0 + 1) * LDS_PITCH + ic + c];
                S = __builtin_amdgcn_wmma_f32_16x16x4_f32(
                        false, a, false, bj[kk], (short)0, S, false, false);
            }

            // ---- P = exp(BW*S) = exp2(S * BW*log2e); accumulate column degrees; swapped copy
            float p[8], q[8];
#pragma unroll
            for (int r = 0; r < 8; ++r) {
                p[r] = __builtin_amdgcn_exp2f(S[r] * (BW_F * LOG2E_F));
                deg += p[r];
                q[r] = swap16(p[r]);
            }

            // ---- gemm2: Y(32d x 16j) += Xpanel(32d x 16i) * P(16i x 16j), 4 K-chunks of 4
#pragma unroll
            for (int kk = 0; kk < 4; ++kk) {
                v2f b;
                b.x = SEL_B(4 * kk + 0, 4 * kk + 2);
                b.y = SEL_B(4 * kk + 1, 4 * kk + 3);
                const int rr = lo ? lane : lane - 16;     // A-fragment M index
                const int k0 = ic + 4 * kk + (lo ? 0 : 2);
                const int k1 = ic + 4 * kk + (lo ? 1 : 3);
                v2f a0, a1;
                a0.x = L[(rr +  0) * LDS_PITCH + k0];
                a0.y = L[(rr +  0) * LDS_PITCH + k1];
                a1.x = L[(rr + 16) * LDS_PITCH + k0];
                a1.y = L[(rr + 16) * LDS_PITCH + k1];
                acc0 = __builtin_amdgcn_wmma_f32_16x16x4_f32(
                           false, a0, false, b, (short)0, acc0, false, false);
                acc1 = __builtin_amdgcn_wmma_f32_16x16x4_f32(
                           false, a1, false, b, (short)0, acc1, false, false);
            }
        }
        __syncthreads();   // everyone done reading buf[t&1] before overwriting it

        if (t + 2 < NPANELS) issue_panel((t + 2) * PANEL, t & 1);
    }

    // ---- finalize: both lane halves hold partial colsum for col (lane&15)
    const float degT = deg + swap16(deg);
    const float inv  = 1.0f / degT;

#pragma unroll
    for (int r = 0; r < 8; ++r) {
        const int d0   = r + (lo ? 0 : 8);
        const int idx0 = d0 * NPIX + jbase + c;
        Xout[idx0] = ETA_F * (acc0[r] * inv) + (1.0f - ETA_F) * Xin[idx0];
        const int d1   = d0 + 16;
        const int idx1 = d1 * NPIX + jbase + c;
        Xout[idx1] = ETA_F * (acc1[r] * inv) + (1.0f - ETA_F) * Xin[idx1];
    }
}

__global__ __launch_bounds__(256)
void ms_copy_kernel(const float4* __restrict__ src, float4* __restrict__ dst) {
    const int i = blockIdx.x * blockDim.x + threadIdx.x;
    dst[i] = src[i];
}

extern "C" void kernel_launch(void* const* d_in, const int* in_sizes, int n_in,
                              void* d_out, int out_size, void* d_ws, size_t ws_size,
                              hipStream_t stream) {
    (void)in_sizes; (void)n_in; (void)out_size; (void)d_ws; (void)ws_size;
    const float* xin = (const float*)d_in[0];
    float* out = (float*)d_out;

    const int slice = EMBED_DIM * NPIX;  // 294912 floats per iterate

    // slice 0 = input iterate
    ms_copy_kernel<<<slice / (4 * 256), 256, 0, stream>>>(
        (const float4*)xin, (float4*)out);

    // 3 mean-shift iterations, each reads slice t, writes slice t+1
    for (int t = 0; t < N_ITERS; ++t) {
        ms_iter_kernel<<<NPIX / JCOLS, BLOCK, 0, stream>>>(
            out + (size_t)t * slice, out + (size_t)(t + 1) * slice);
    }
}